// SwitchRouterLayer_30674656428452
// MI455X (gfx1250) — compile-verified
//
#include <hip/hip_runtime.h>
#include <hip/hip_bf16.h>

// Problem constants (from reference)
#define C_   4
#define T_   4096
#define D_   1024
#define E_   16
#define CAP_ 320
static const size_t CTECAP = (size_t)C_ * T_ * E_ * CAP_;  // 83,886,080

typedef __attribute__((ext_vector_type(2))) float v2f;
typedef __attribute__((ext_vector_type(4))) float v4f;
typedef __attribute__((ext_vector_type(8))) float v8f;

// Workspace layout (bytes):
//  [0,       65536)  gate  : float[16384]
//  [65536,  131072)  eidx  : int[16384]
//  [131072, 196608)  probpart: float[1024 waves][16 experts]  (deterministic partials)
//  [196608, 196864)  cnt   : uint[64]  (C*E atomic counters)
#define WS_GATE   0
#define WS_EIDX   65536
#define WS_PPART  131072
#define WS_CNT    196608

// ---------------------------------------------------------------------------
// Kernel A: stream-zero the 671MB output with nontemporal b128 stores, and
// zero the small atomic-counter region of the workspace.
// ---------------------------------------------------------------------------
__global__ __launch_bounds__(256) void zero_kernel(float* __restrict__ out,
                                                   unsigned* __restrict__ cnt) {
  size_t i = (size_t)blockIdx.x * blockDim.x + threadIdx.x;
  size_t stride = (size_t)gridDim.x * blockDim.x;
  v4f z = {0.f, 0.f, 0.f, 0.f};
  v4f* o4 = (v4f*)out;
  const size_t n4 = (2 * CTECAP) / 4;
  for (size_t j = i; j < n4; j += stride)
    __builtin_nontemporal_store(z, o4 + j);
  if (blockIdx.x == 0) {
    if (threadIdx.x < C_ * E_) cnt[threadIdx.x] = 0u;
    if (threadIdx.x == 0) out[2 * CTECAP] = 0.0f;  // aux slot (rewritten later)
  }
}

// ---------------------------------------------------------------------------
// Kernel B: router logits GEMM via V_WMMA_F32_16X16X4_F32 + softmax/argmax.
// One wave = one 16-token x 16-expert tile. 1024 waves total.
//
// A (16x4 f32) lane layout (ISA 7.12.2): lanes 0-15 hold M=lane, {K=0,K=1};
// lanes 16-31 hold M=lane-16, {K=2,K=3} -> per-lane contiguous float2 load.
// B (4x16, = W^T chunk) mirrored: per-lane float2 of W[row=lane&15][k+2*half].
// C/D (16x16 f32): lane holds expert N=lane&15, tokens M = 8*(lane>>4)+v.
// ---------------------------------------------------------------------------
__global__ __launch_bounds__(256) void router_gemm_kernel(
    const float* __restrict__ inp, const float* __restrict__ wgt,
    const float* __restrict__ nse, float* __restrict__ gate,
    int* __restrict__ eidx, float* __restrict__ probpart,
    unsigned* __restrict__ cnt) {
  const int lane = threadIdx.x & 31;
  const int ln   = lane & 15;
  const int h    = lane >> 4;
  const int wave = blockIdx.x * (blockDim.x >> 5) + (threadIdx.x >> 5);
  const int r0   = wave * 16;  // first token row of this tile

  const float* Ab = inp + (size_t)(r0 + ln) * D_;
  const float* Nb = nse + (size_t)(r0 + ln) * D_;
  const float* Bb = wgt + (size_t)ln * D_;  // router_weight row = expert ln

  v8f acc = {0.f, 0.f, 0.f, 0.f, 0.f, 0.f, 0.f, 0.f};
#pragma unroll 4
  for (int k = 0; k < D_; k += 4) {
    const int kk = k + 2 * h;
    v2f xa = *(const v2f*)(Ab + kk);
    v2f xn = *(const v2f*)(Nb + kk);
    v2f a  = xa * xn;                      // jittered gate input
    v2f b  = *(const v2f*)(Bb + kk);
    acc = __builtin_amdgcn_wmma_f32_16x16x4_f32(
        /*neg_a=*/false, a, /*neg_b=*/false, b,
        /*c_mod=*/(short)0, acc, /*reuse_a=*/false, /*reuse_b=*/false);
  }

  // Softmax / argmax over E=16 (across lanes within each half-wave).
  float probsum_e = 0.f;   // this lane's expert prob summed over its 8 tokens
  unsigned mycnt  = 0u;    // tokens whose argmax == this lane's expert
  float gates[8];
  int   idxs[8];
#pragma unroll
  for (int v = 0; v < 8; ++v) {
    float lv = acc[v];
    float mval = lv;
    int   midx = ln;
#pragma unroll
    for (int m = 8; m >= 1; m >>= 1) {
      float ov = __shfl_xor(mval, m, 16);
      int   oi = __shfl_xor(midx, m, 16);
      if (ov > mval || (ov == mval && oi < midx)) { mval = ov; midx = oi; }
    }
    float e = expf(lv - mval);
    float s = e;
#pragma unroll
    for (int m = 8; m >= 1; m >>= 1) s += __shfl_xor(s, m, 16);
    probsum_e += e / s;
    mycnt     += (midx == ln) ? 1u : 0u;
    gates[v] = 1.0f / s;  // max prob = exp(mval - mval) / s
    idxs[v]  = midx;
  }

  // Deterministic per-wave probability partials: fold the two halves, lane
  // ln in half 0 writes the 16-token partial for expert ln.
  float pall = probsum_e + __shfl_xor(probsum_e, 16, 32);
  if (h == 0) probpart[(size_t)wave * 16 + ln] = pall;

  // Integer atomics are order-independent -> deterministic.
  const int cbase = (r0 >> 12) << 4;  // c * 16
  atomicAdd(&cnt[cbase + ln], mycnt);

  if (ln == 0) {
#pragma unroll
    for (int v = 0; v < 8; ++v) {
      const int row = r0 + h * 8 + v;
      gate[row] = gates[v];
      eidx[row] = idxs[v];
    }
  }
}

// ---------------------------------------------------------------------------
// Kernel C: aux loss + the (at most 64) sparse nonzero output entries.
// tid = c*16 + j ; j doubles as expert-id (hitset test) and token position.
// ---------------------------------------------------------------------------
__global__ __launch_bounds__(64) void finalize_kernel(
    const float* __restrict__ gate, const int* __restrict__ eidx,
    const float* __restrict__ probpart, const unsigned* __restrict__ cnt,
    float* __restrict__ out) {
  __shared__ float auxp[64];
  const int tid = threadIdx.x;
  const int c = tid >> 4;
  const int j = tid & 15;

  // density1_proxy * density1 for (c, e=j): deterministic serial sum over the
  // 256 wave-partials belonging to chip c.
  float ps = 0.f;
  const float* pp = probpart + (size_t)c * 256 * 16;
  for (int w = 0; w < 256; ++w) ps += pp[w * 16 + j];
  auxp[tid] = ((float)cnt[tid] / (float)T_) * (ps / (float)T_);
  __syncthreads();
  if (tid == 0) {
    float s = 0.f;
    for (int i = 0; i < C_ * E_; ++i) s += auxp[i];
    out[2 * CTECAP] = s * (float)E_;
  }

  // Sparse writes: token t0=j is "hit" iff expert id j was chosen by any token
  // in chip c (count > 0). Tokens t0<16 always survive capacity (rank<=16<320).
  if (cnt[c * 16 + j] > 0u) {
    const int* er = eidx + (size_t)c * T_;
    const int et = er[j];
    int rank = 0;
    for (int t = 0; t <= j; ++t) rank += (er[t] == et) ? 1 : 0;
    const int p = (et == 0) ? rank : 0;  // position_in_expert[c,t0,e=0]
    const size_t base = (((size_t)c * T_ + j) * E_ + 0) * CAP_ + p;
    out[base] = 1.0f;                                   // dispatch (bool)
    out[CTECAP + base] = gate[(size_t)c * T_ + j];      // combine
  }
}

// ---------------------------------------------------------------------------
extern "C" void kernel_launch(void* const* d_in, const int* in_sizes, int n_in,
                              void* d_out, int out_size, void* d_ws,
                              size_t ws_size, hipStream_t stream) {
  (void)in_sizes; (void)n_in; (void)out_size; (void)ws_size;
  const float* inp = (const float*)d_in[0];  // inputs        [C,T,D] f32
  const float* wgt = (const float*)d_in[1];  // router_weight [E,D]   f32
  const float* nse = (const float*)d_in[2];  // noise         [C,T,D] f32
  float* out = (float*)d_out;

  char* ws = (char*)d_ws;
  float*    gate     = (float*)(ws + WS_GATE);
  int*      eidx     = (int*)(ws + WS_EIDX);
  float*    probpart = (float*)(ws + WS_PPART);
  unsigned* cnt      = (unsigned*)(ws + WS_CNT);

  zero_kernel<<<2048, 256, 0, stream>>>(out, cnt);
  router_gemm_kernel<<<128, 256, 0, stream>>>(inp, wgt, nse, gate, eidx,
                                              probpart, cnt);
  finalize_kernel<<<1, 64, 0, stream>>>(gate, eidx, probpart, cnt, out);
}